// DySample_74526272520677
// MI455X (gfx1250) — compile-verified
//
#include <hip/hip_runtime.h>

// ---------------------------------------------------------------------------
// DySample (scale=2, groups=4) for MI455X / gfx1250.
// Kernel 1: offset = W_off(32x256) @ x per pixel via V_WMMA_F32_16X16X4_F32,
//           dual accumulator chains, async global->LDS staging when available.
// Kernel 2: bilinear border grid-sample; b128 stores, prefetch, 134 MB stream.
// ---------------------------------------------------------------------------

typedef float v2f __attribute__((ext_vector_type(2)));
typedef float v8f __attribute__((ext_vector_type(8)));
typedef int   v4i __attribute__((vector_size(16)));
// Pointer types matching the async-to-LDS builtin's signature:
// arg0: v4i in AS1 (global, printed as "__device__" by clang in HIP mode),
// arg1: v4i in AS3 (LDS).
typedef __attribute__((address_space(1))) v4i* gv4i_p;
typedef __attribute__((address_space(3))) v4i* lv4i_p;

#define B_      8
#define C_      256
#define H_      64
#define W_      64
#define HW_     (H_ * W_)          // 4096
#define OCH_    32                 // 2 * groups * scale * scale
#define LDSW    40                 // row stride (floats): 2*40 % 64 == 16 -> lane
                                   // halves hit disjoint LDS bank ranges

#if __has_builtin(__builtin_amdgcn_global_load_async_to_lds_b128) && \
    __has_builtin(__builtin_amdgcn_s_wait_asynccnt)
#define HAVE_ASYNC_LDS 1
#else
#define HAVE_ASYNC_LDS 0
#endif

// --------------------------- Kernel 1: offset GEMM --------------------------
// Block = 128 threads (4 waves). Block owns 32 pixels of one batch image.
// wave w: pixel-tile t = w&1 (16 pixels), m-tile mt = w>>1 (out chans 16*mt..).
// Two independent WMMA accumulator chains hide the D->C RAW latency.
__global__ __launch_bounds__(128) void dysample_offset_gemm(
    const float* __restrict__ x, const float* __restrict__ Woff,
    float* __restrict__ ofs)
{
    __shared__ float wsT[C_ * LDSW];   // W_off transposed: wsT[k*LDSW + m]
    __shared__ float xs [C_ * LDSW];   // x tile:           xs [c*LDSW + p]

    const int tid  = threadIdx.x;
    const int base = blockIdx.x * 32;          // flat pixel base
    const int b    = base >> 12;               // / 4096
    const int p0   = base & (HW_ - 1);

    // Stage W_off (32x256, row-major) transposed into LDS (scatter, one-time).
    const float4* w4 = reinterpret_cast<const float4*>(Woff);
    for (int idx = tid; idx < (OCH_ * C_) / 4; idx += 128) {
        float4 v = w4[idx];                    // coalesced b128 global loads
        int m = idx >> 6;                      // 64 float4 per 256-wide row
        int k = (idx & 63) * 4;
        wsT[(k + 0) * LDSW + m] = v.x;
        wsT[(k + 1) * LDSW + m] = v.y;
        wsT[(k + 2) * LDSW + m] = v.z;
        wsT[(k + 3) * LDSW + m] = v.w;
    }
    // Stage 32-pixel x tile (all 256 channels), contiguous along pixels.
    // Prefer the CDNA5 async global->LDS path (ASYNCcnt) when the toolchain
    // exposes it; otherwise classic b128 load + ds_store.
    const float* xb = x + (size_t)b * (C_ * HW_) + p0;
    for (int idx = tid; idx < C_ * 8; idx += 128) {
        int cch = idx >> 3;                    // channel
        int q   = idx & 7;                     // float4 slot within 32 pixels
        const float* src = xb + cch * HW_ + q * 4;
        float*       dst = &xs[cch * LDSW + q * 4];
#if HAVE_ASYNC_LDS
        __builtin_amdgcn_global_load_async_to_lds_b128(
            (gv4i_p)(void*)src, (lv4i_p)(void*)dst,
            /*offset=*/0, /*cpol=*/0);
#else
        *reinterpret_cast<float4*>(dst) = *reinterpret_cast<const float4*>(src);
#endif
    }
#if HAVE_ASYNC_LDS
    __builtin_amdgcn_s_wait_asynccnt(0);
#endif
    __syncthreads();

    const int wave = tid >> 5, lane = tid & 31;
    const int t  = wave & 1;                   // pixel sub-tile
    const int mt = wave >> 1;                  // output-channel tile
    const int hi = lane >> 4, lo = lane & 15;
    const int m   = mt * 16 + lo;              // A row (out channel)
    const int col = t * 16 + lo;               // B column (pixel)
    const int kb  = hi * 2;                    // lane-half K skew (ISA layout)

    v8f acc0 = {0.f, 0.f, 0.f, 0.f, 0.f, 0.f, 0.f, 0.f};
    v8f acc1 = {0.f, 0.f, 0.f, 0.f, 0.f, 0.f, 0.f, 0.f};
    #pragma unroll 4
    for (int k0 = 0; k0 < C_; k0 += 8) {
        v2f a0, b0, a1, b1;
        // A 16x4: VGPR0 = K even (k0 / k0+2 per half), VGPR1 = K odd.
        a0.x = wsT[(k0 + kb + 0) * LDSW + m];
        a0.y = wsT[(k0 + kb + 1) * LDSW + m];
        b0.x = xs [(k0 + kb + 0) * LDSW + col];
        b0.y = xs [(k0 + kb + 1) * LDSW + col];
        a1.x = wsT[(k0 + 4 + kb + 0) * LDSW + m];
        a1.y = wsT[(k0 + 4 + kb + 1) * LDSW + m];
        b1.x = xs [(k0 + 4 + kb + 0) * LDSW + col];
        b1.y = xs [(k0 + 4 + kb + 1) * LDSW + col];
        acc0 = __builtin_amdgcn_wmma_f32_16x16x4_f32(
            false, a0, false, b0, (short)0, acc0, false, false);
        acc1 = __builtin_amdgcn_wmma_f32_16x16x4_f32(
            false, a1, false, b1, (short)0, acc1, false, false);
    }
    v8f acc = acc0 + acc1;

    // C/D layout: VGPR r -> M = r + 8*hi, N = lane&15.
    float* outp = ofs + (size_t)b * (OCH_ * HW_) + p0 + t * 16 + lo;
    #pragma unroll
    for (int r = 0; r < 8; ++r)
        outp[(mt * 16 + hi * 8 + r) * HW_] = acc[r];
}

// --------------------------- Kernel 2: grid sample ---------------------------
// Grid math collapses to: ix = w + offx, iy = h + offy with
// offx = 0.25*(raw_x + b_off) + (c-0.5)/2 (b_off and _init_pos folded here).
// Block = 128 threads = 4 waves; each wave owns one output row (b,g,Y);
// each lane owns 4 consecutive X -> aligned b128 stores, 16 gathers/channel.
__global__ __launch_bounds__(128) void dysample_sample(
    const float* __restrict__ x, const float* __restrict__ ofs,
    const float* __restrict__ boff, float* __restrict__ out)
{
    const int lane = threadIdx.x & 31;
    const int yq   = threadIdx.x >> 5;          // row within block's 4 rows
    int id = blockIdx.x;
    const int Y = (id & 31) * 4 + yq;  id >>= 5; // 128 rows / 4 per block
    const int g = id & 3;              id >>= 2;
    const int b = id;

    const int h = Y >> 1, r = Y & 1;
    const size_t obase = (size_t)b * (OCH_ * HW_);

    float wgt[4][4];
    int   cidx[4][4];
    #pragma unroll
    for (int j = 0; j < 4; ++j) {
        const int X = lane * 4 + j;
        const int w = X >> 1, c = X & 1;
        const int p  = r * 2 + c;
        const int ox = g * 4 + p;               // offset channel, x (d=0)
        const int oy = 16 + g * 4 + p;          // offset channel, y (d=1)
        const int pix = h * W_ + w;
        const float rawx = ofs[obase + (size_t)ox * HW_ + pix];
        const float rawy = ofs[obase + (size_t)oy * HW_ + pix];
        const float offx = 0.25f * (rawx + boff[ox]) + ((float)c - 0.5f) * 0.5f;
        const float offy = 0.25f * (rawy + boff[oy]) + ((float)r - 0.5f) * 0.5f;

        const float ix = (float)w + offx;
        const float iy = (float)h + offy;
        const float x0f = floorf(ix), y0f = floorf(iy);
        const float wx = ix - x0f,    wy = iy - y0f;
        const int x0 = min(max((int)x0f,     0), W_ - 1);
        const int x1 = min(max((int)x0f + 1, 0), W_ - 1);
        const int y0 = min(max((int)y0f,     0), H_ - 1);
        const int y1 = min(max((int)y0f + 1, 0), H_ - 1);

        wgt[j][0] = (1.f - wx) * (1.f - wy);
        wgt[j][1] = wx * (1.f - wy);
        wgt[j][2] = (1.f - wx) * wy;
        wgt[j][3] = wx * wy;
        cidx[j][0] = y0 * W_ + x0;  cidx[j][1] = y0 * W_ + x1;
        cidx[j][2] = y1 * W_ + x0;  cidx[j][3] = y1 * W_ + x1;
    }

    const float* img = x + ((size_t)b * C_ + g * 64) * HW_;
    float* op = out + (((size_t)b * C_ + g * 64) * 128 + Y) * 128 + lane * 4;
    #pragma unroll 2
    for (int ch = 0; ch < 64; ++ch) {
        const float* im = img + (size_t)ch * HW_;
        // Prefetch 8 channels ahead (clamped so the address stays in-bounds).
        const int chp = (ch + 8 < 64) ? ch + 8 : 63;
        __builtin_prefetch(img + (size_t)chp * HW_ + cidx[0][0], 0, 0);

        float vv[4];
        #pragma unroll
        for (int j = 0; j < 4; ++j)
            vv[j] = im[cidx[j][0]] * wgt[j][0] + im[cidx[j][1]] * wgt[j][1]
                  + im[cidx[j][2]] * wgt[j][2] + im[cidx[j][3]] * wgt[j][3];
        float4 v = {vv[0], vv[1], vv[2], vv[3]};
        *reinterpret_cast<float4*>(op + (size_t)ch * (128 * 128)) = v;
    }
}

// ------------------------------------ launch ---------------------------------
extern "C" void kernel_launch(void* const* d_in, const int* in_sizes, int n_in,
                              void* d_out, int out_size, void* d_ws, size_t ws_size,
                              hipStream_t stream) {
    const float* x    = (const float*)d_in[0];   // (8,256,64,64) fp32
    const float* Woff = (const float*)d_in[1];   // (32,256)      fp32
    const float* boff = (const float*)d_in[2];   // (32,)         fp32
    float*       out  = (float*)d_out;           // (8,256,128,128)
    float*       ofs  = (float*)d_ws;            // 8*32*64*64 fp32 = 4 MB

    // Kernel 1: 32768 pixels / 32 per block = 1024 blocks.
    dysample_offset_gemm<<<(B_ * HW_) / 32, 128, 0, stream>>>(x, Woff, ofs);
    // Kernel 2: (b, g, 4-row slab): 8*4*32 = 1024 blocks x 128 threads.
    dysample_sample<<<B_ * 4 * 32, 128, 0, stream>>>(x, ofs, boff, out);
}